// Tucker_41884521071233
// MI455X (gfx1250) — compile-verified
//
#include <hip/hip_runtime.h>
#include <hip/hip_bf16.h>

// CDNA5 / gfx1250 fused Tucker-style trilinear scoring kernel.
// score[b] = sum_{i,t,r} head[b,i] * tail[b,t] * W[i,t,r] * p[b,r]
// Dominant GEMM (68.7 GFLOP fp32) on V_WMMA_F32_16X16X4_F32.
// v4: WGP-scope prefetch via inline asm (default scope 0 = WGP: fills all
//     cache levels), replacing the useless SE/SYS-scope builtin prefetch.

typedef float v2f __attribute__((ext_vector_type(2)));
typedef float v8f __attribute__((ext_vector_type(8)));

#define BTOT 2048
#define DDIM 256
#define RDIM 256
#define LDS_STRIDE 264   // padded i-stride (dwords) for the [r][i] W tile
#define T_CHUNK 32       // t tiles per workgroup (grid.z = 256 / T_CHUNK)

__global__ __launch_bounds__(256) void tucker_wmma_f32_kernel(
    const float* __restrict__ p,   // (B, R)
    const float* __restrict__ c,   // (B, 2, D)  [:,0,:]=head  [:,1,:]=tail
    const float* __restrict__ W,   // (D, D, R)  row-major: W[i,t,r]
    float* __restrict__ out)       // (B,) — pre-zeroed, accumulated via atomics
{
    // Double-buffered W tile W[:, t, r0:r0+16], transposed: ldsW[buf][r_local][i].
    __shared__ float ldsW[2][16 * LDS_STRIDE];   // 2 x 16.9 KB

    const int lane  = threadIdx.x & 31;
    const int wave  = threadIdx.x >> 5;              // 0..7 -> batch group in cluster
    const int g16   = (blockIdx.x * 8 + wave) * 16;  // first batch of this group
    const int r0    = blockIdx.y * 16;               // r block
    const int t_beg = blockIdx.z * T_CHUNK;          // t chunk
    const int t_end = t_beg + T_CHUNK;

    const int mrow = lane & 15;   // A: M row | B/C: N col
    const int hi   = lane >> 4;   // half-wave selector
    const int koff = hi * 2;      // A/B fragment K offset (K=0,1 vs K=2,3)

    const int i_idx = threadIdx.x;                 // 0..255 == i index for W loads
    const float* wbase = W + (size_t)i_idx * (DDIM * RDIM) + r0;  // + t*RDIM per tile

    // ---- A (head) fragments, K = 256 in 64 steps of 4: resident in VGPRs ----
    v2f afrag[64];
    {
        const float* hrow = c + (size_t)(g16 + mrow) * (2 * DDIM);  // head row
        #pragma unroll
        for (int s = 0; s < 64; ++s)
            afrag[s] = *(const v2f*)(hrow + 4 * s + koff);
    }

    // Persistent 16x16 tile of W2[b, r0+n] partials (summed over this t chunk)
    v8f acc2 = {0.f, 0.f, 0.f, 0.f, 0.f, 0.f, 0.f, 0.f};

    // ---- prologue: stage tile t_beg into ldsW[0] ----
    {
        const float* src = wbase + (size_t)t_beg * RDIM;
        #pragma unroll
        for (int q = 0; q < 4; ++q) {
            float4 wq = *(const float4*)(src + 4 * q);
            ldsW[0][(4 * q + 0) * LDS_STRIDE + i_idx] = wq.x;
            ldsW[0][(4 * q + 1) * LDS_STRIDE + i_idx] = wq.y;
            ldsW[0][(4 * q + 2) * LDS_STRIDE + i_idx] = wq.z;
            ldsW[0][(4 * q + 3) * LDS_STRIDE + i_idx] = wq.w;
        }
    }
    __syncthreads();

    int buf = 0;
    for (int t = t_beg; t < t_end; ++t) {
        const bool have_next = (t + 1 < t_end);

        // ---- issue global loads for tile t+1 (latency hidden under WMMAs) ----
        float4 nxt0, nxt1, nxt2, nxt3;
        if (have_next) {
            const float* src = wbase + (size_t)(t + 1) * RDIM;
            nxt0 = *(const float4*)(src + 0);
            nxt1 = *(const float4*)(src + 4);
            nxt2 = *(const float4*)(src + 8);
            nxt3 = *(const float4*)(src + 12);
            if (t + 2 < t_end) {
                // WGP-scope prefetch (scope 0: fill all cache levels) for t+2
                const float* pf = wbase + (size_t)(t + 2) * RDIM;
                asm volatile("global_prefetch_b8 %0, off" :: "v"(pf));
            }
        }

        // ---- W1 tile: (16 batches) x (16 r), K=256 over i ----
        v8f cacc = {0.f, 0.f, 0.f, 0.f, 0.f, 0.f, 0.f, 0.f};
        const float* bbase = &ldsW[buf][mrow * LDS_STRIDE + koff];
        #pragma unroll
        for (int s = 0; s < 64; ++s) {
            v2f bfrag = *(const v2f*)(bbase + 4 * s);
            cacc = __builtin_amdgcn_wmma_f32_16x16x4_f32(
                false, afrag[s], false, bfrag, (short)0, cacc, false, false);
        }

        // ---- fold tail[b, t] into W2 accumulator (row M = k + 8*hi) ----
        const float* tailb = c + (size_t)(g16 + 8 * hi) * (2 * DDIM) + DDIM + t;
        #pragma unroll
        for (int k = 0; k < 8; ++k) {
            float tl = tailb[(size_t)k * (2 * DDIM)];
            acc2[k] += tl * cacc[k];
        }

        // ---- drain staged tile t+1 into the other LDS buffer ----
        if (have_next) {
            float* dst = ldsW[buf ^ 1];
            dst[( 0) * LDS_STRIDE + i_idx] = nxt0.x;
            dst[( 1) * LDS_STRIDE + i_idx] = nxt0.y;
            dst[( 2) * LDS_STRIDE + i_idx] = nxt0.z;
            dst[( 3) * LDS_STRIDE + i_idx] = nxt0.w;
            dst[( 4) * LDS_STRIDE + i_idx] = nxt1.x;
            dst[( 5) * LDS_STRIDE + i_idx] = nxt1.y;
            dst[( 6) * LDS_STRIDE + i_idx] = nxt1.z;
            dst[( 7) * LDS_STRIDE + i_idx] = nxt1.w;
            dst[( 8) * LDS_STRIDE + i_idx] = nxt2.x;
            dst[( 9) * LDS_STRIDE + i_idx] = nxt2.y;
            dst[(10) * LDS_STRIDE + i_idx] = nxt2.z;
            dst[(11) * LDS_STRIDE + i_idx] = nxt2.w;
            dst[(12) * LDS_STRIDE + i_idx] = nxt3.x;
            dst[(13) * LDS_STRIDE + i_idx] = nxt3.y;
            dst[(14) * LDS_STRIDE + i_idx] = nxt3.z;
            dst[(15) * LDS_STRIDE + i_idx] = nxt3.w;
        }

        __syncthreads();   // readers of ldsW[buf] done; writes to buf^1 visible
        buf ^= 1;
    }

    // ---- multiply by p[b, r0+N], reduce over N within each 16-lane half ----
    #pragma unroll
    for (int k = 0; k < 8; ++k) {
        const int b = g16 + k + 8 * hi;
        float v = acc2[k] * p[(size_t)b * RDIM + r0 + mrow];
        v += __shfl_xor(v, 1, 32);
        v += __shfl_xor(v, 2, 32);
        v += __shfl_xor(v, 4, 32);
        v += __shfl_xor(v, 8, 32);
        if (mrow == 0) atomicAdd(&out[b], v);
    }
}

extern "C" void kernel_launch(void* const* d_in, const int* in_sizes, int n_in,
                              void* d_out, int out_size, void* d_ws, size_t ws_size,
                              hipStream_t stream) {
    const float* p = (const float*)d_in[0];   // (2048, 256)
    const float* c = (const float*)d_in[1];   // (2048, 2, 256)
    const float* W = (const float*)d_in[2];   // (256, 256, 256)
    float* out = (float*)d_out;               // (2048, 1)

    hipMemsetAsync(out, 0, (size_t)out_size * sizeof(float), stream);

    dim3 grid(16, 16, 256 / T_CHUNK);  // batch clusters x r-blocks x t-chunks
    dim3 block(256);                   // 8 waves; wave w owns group cluster*8+w
    tucker_wmma_f32_kernel<<<grid, block, 0, stream>>>(p, c, W, out);
}